// RoIDelta_40157944217902
// MI455X (gfx1250) — compile-verified
//
#include <hip/hip_runtime.h>
#include <cstdint>

// Problem constants (from reference)
#define B_   64
#define N_   1500
#define M_   100
#define MP_  112        // gt count padded to 7 tiles of 16
#define L_   81
#define SEL_ 128
#define ROI_PER_BLOCK 128
#define CHUNKS 12       // ceil(1500/128)

typedef __attribute__((ext_vector_type(2))) float v2f;
typedef __attribute__((ext_vector_type(8))) float v8f;

// ---------------------------------------------------------------------------
// Threefry2x32 (matches JAX PRNG), usable host+device.
// ---------------------------------------------------------------------------
__host__ __device__ __forceinline__ uint32_t rotl32(uint32_t v, int r) {
  return (v << r) | (v >> (32 - r));
}
__host__ __device__ inline void threefry2x32(uint32_t k0, uint32_t k1,
                                             uint32_t x0, uint32_t x1,
                                             uint32_t& o0, uint32_t& o1) {
  const uint32_t ks2 = 0x1BD11BDAu ^ k0 ^ k1;
  const int RA[4] = {13, 15, 26, 6};
  const int RB[4] = {17, 29, 16, 24};
  x0 += k0; x1 += k1;
  for (int i = 0; i < 4; ++i) { x0 += x1; x1 = rotl32(x1, RA[i]); x1 ^= x0; }
  x0 += k1; x1 += ks2 + 1u;
  for (int i = 0; i < 4; ++i) { x0 += x1; x1 = rotl32(x1, RB[i]); x1 ^= x0; }
  x0 += ks2; x1 += k0 + 2u;
  for (int i = 0; i < 4; ++i) { x0 += x1; x1 = rotl32(x1, RA[i]); x1 ^= x0; }
  x0 += k0; x1 += k1 + 3u;
  for (int i = 0; i < 4; ++i) { x0 += x1; x1 = rotl32(x1, RB[i]); x1 ^= x0; }
  x0 += k1; x1 += ks2 + 4u;
  for (int i = 0; i < 4; ++i) { x0 += x1; x1 = rotl32(x1, RA[i]); x1 ^= x0; }
  x0 += ks2; x1 += k0 + 5u;
  o0 = x0; o1 = x1;
}

__device__ __forceinline__ float bits_to_u01(uint32_t bits) {
  // JAX: bitcast((bits >> 9) | 0x3f800000) - 1.0
  return __uint_as_float((bits >> 9) | 0x3F800000u) - 1.0f;
}

// ---------------------------------------------------------------------------
// CDNA5 async global->LDS copy (ASYNCcnt path). Builtin signature is
// (global int* src, shared int* dst, imm offset, imm cpol) per hipcc Sema.
// LDS flat-pointer low 32 bits == LDS offset (ISA §10.2 aperture rules).
// ---------------------------------------------------------------------------
__device__ __forceinline__ void async_copy_b32(float* lds_dst, const float* g_src) {
#if __has_builtin(__builtin_amdgcn_global_load_async_to_lds_b32)
  __builtin_amdgcn_global_load_async_to_lds_b32(
      (__attribute__((address_space(1))) int*)(unsigned long long)g_src,
      (__attribute__((address_space(3))) int*)(unsigned int)(unsigned long long)lds_dst,
      0, 0);
#else
  uint32_t laddr = (uint32_t)(unsigned long long)lds_dst;
  asm volatile("global_load_async_to_lds_b32 %0, %1, off"
               :: "v"(laddr), "v"(g_src) : "memory");
#endif
}
__device__ __forceinline__ void wait_async0() {
#if __has_builtin(__builtin_amdgcn_s_wait_asynccnt)
  __builtin_amdgcn_s_wait_asynccnt(0);
#else
  asm volatile("s_wait_asynccnt 0x0" ::: "memory");
#endif
}

// ---------------------------------------------------------------------------
// Kernel A: per-block = one batch row chunk of 128 rois.
// Stage gt boxes (AoS->SoA) + roi boxes into LDS with async copies.
// Each wave owns 16 rois; loops 7 gt tiles of 16.
// Union broadcast (area_b[i] + area_g[j]) computed on the matrix pipe with
// V_WMMA_F32_16X16X4_F32 (A = [area_b, 1, 0, 0] 16x4; B = [1; area_g; 0; 0]
// 4x16), co-executing with the VALU intersection min/max math.
// ---------------------------------------------------------------------------
__global__ __launch_bounds__(256) void iou_argmax_kernel(
    const float* __restrict__ roi, const float* __restrict__ gt,
    float* __restrict__ mergedOut, int* __restrict__ maxIdxOut) {
  __shared__ float gtS[4][MP_];     // SoA y1,x1,y2,x2
  __shared__ float gtArea[MP_];
  __shared__ float roiS[4][ROI_PER_BLOCK];

  const int b = blockIdx.x / CHUNKS;
  const int chunk = blockIdx.x % CHUNKS;
  const int roiBase = chunk * ROI_PER_BLOCK;
  const int tid = threadIdx.x;

  // zero the gt padding slots [100,112)
  if (tid < (MP_ - M_) * 4) {
    gtS[tid & 3][M_ + (tid >> 2)] = 0.0f;
  }
  // async-scatter gt AoS -> LDS SoA (400 b32 transfers)
  const float* gtb = gt + (size_t)b * M_ * 4;
  for (int e = tid; e < M_ * 4; e += 256) {
    async_copy_b32(&gtS[e & 3][e >> 2], gtb + e);
  }
  // async-scatter this block's 128 roi boxes -> LDS SoA (clamped tail)
  const float* roib = roi + (size_t)b * N_ * 4;
  for (int e = tid; e < ROI_PER_BLOCK * 4; e += 256) {
    int r = e >> 2, c = e & 3;
    int rg = roiBase + r; if (rg > N_ - 1) rg = N_ - 1;
    async_copy_b32(&roiS[c][r], roib + rg * 4 + c);
  }
  wait_async0();
  __syncthreads();

  if (tid < MP_) {
    gtArea[tid] = (gtS[2][tid] - gtS[0][tid]) * (gtS[3][tid] - gtS[1][tid]);
  }
  __syncthreads();

  const int lane = tid & 31;
  const int wave = tid >> 5;
  const int hi = lane >> 4;      // 0: D rows M=0..7, 1: M=8..15
  const int lj = lane & 15;      // gt column within tile / roi index for A
  const int wOff = wave * 16;    // this wave's roi tile offset within block

  // A matrix 16x4: VGPR0 lanes0-15 = K0 (area_b), lanes16-31 = K2 (0);
  //               VGPR1 lanes0-15 = K1 (1.0),    lanes16-31 = K3 (0).
  float abh = roiS[2][wOff + lj] - roiS[0][wOff + lj];
  float abw = roiS[3][wOff + lj] - roiS[1][wOff + lj];
  v2f amat = { hi ? 0.0f : abh * abw, hi ? 0.0f : 1.0f };

  // hoist the 8 roi rows this lane accumulates (tile-invariant)
  float ry1[8], rx1[8], ry2[8], rx2[8];
#pragma unroll
  for (int v = 0; v < 8; ++v) {
    int m = wOff + v + 8 * hi;
    ry1[v] = roiS[0][m]; rx1[v] = roiS[1][m];
    ry2[v] = roiS[2][m]; rx2[v] = roiS[3][m];
  }

  float best[8]; int bestJ[8];
#pragma unroll
  for (int v = 0; v < 8; ++v) { best[v] = -1.0f; bestJ[v] = 0; }

  for (int t = 0; t < MP_ / 16; ++t) {
    const int jj = t * 16 + lj;
    const float gy1 = gtS[0][jj], gx1 = gtS[1][jj];
    const float gy2 = gtS[2][jj], gx2 = gtS[3][jj];
    const float gA  = gtArea[jj];
    // B matrix 4x16: VGPR0 lanes0-15 = row K0 (1.0), VGPR1 lanes0-15 = row K1
    // (area_g); lanes16-31 carry rows K2/K3 (0).
    v2f bmat = { hi ? 0.0f : 1.0f, hi ? 0.0f : gA };
    v8f cz = {0.f, 0.f, 0.f, 0.f, 0.f, 0.f, 0.f, 0.f};
    // D[m][j] = area_b[m] + area_g[j]   (matrix pipe; co-executes with VALU)
    v8f d = __builtin_amdgcn_wmma_f32_16x16x4_f32(
        false, amat, false, bmat, (short)0, cz, false, false);
#pragma unroll
    for (int v = 0; v < 8; ++v) {
      float iy = fminf(ry2[v], gy2) - fmaxf(ry1[v], gy1);
      float ix = fminf(rx2[v], gx2) - fmaxf(rx1[v], gx1);
      float inter = fmaxf(iy, 0.0f) * fmaxf(ix, 0.0f);
      float iou = inter / (d[v] - inter + 1e-7f);
      if (iou > best[v]) { best[v] = iou; bestJ[v] = jj; }  // keeps earliest j
    }
  }

  // stable argmax reduction across the 16 lanes of each half (butterfly)
#pragma unroll
  for (int off = 1; off < 16; off <<= 1) {
#pragma unroll
    for (int v = 0; v < 8; ++v) {
      float ov = __shfl_xor(best[v], off, 32);
      int   oj = __shfl_xor(bestJ[v], off, 32);
      if (ov > best[v] || (ov == best[v] && oj < bestJ[v])) {
        best[v] = ov; bestJ[v] = oj;
      }
    }
  }
#pragma unroll
  for (int v = 0; v < 8; ++v) {
    if (lj == v) {
      int r = roiBase + wOff + v + 8 * hi;
      if (r < N_) {
        size_t g = (size_t)b * N_ + r;
        mergedOut[g] = best[v];
        maxIdxOut[g] = bestJ[v];
      }
    }
  }
}

// ---------------------------------------------------------------------------
// Kernel B: one block per batch row. Threefry scores, pairwise rank count in
// LDS (stable descending sort rank), then matched label + bbox delta.
// ---------------------------------------------------------------------------
__global__ __launch_bounds__(512) void select_delta_kernel(
    const float* __restrict__ roi, const float* __restrict__ gt,
    const int* __restrict__ gtl,
    const float* __restrict__ merged, const int* __restrict__ maxIdx,
    int* __restrict__ labelOut, float* __restrict__ deltaOut,
    uint32_t kp0, uint32_t kp1, uint32_t kn0, uint32_t kn1) {
  __shared__ float sP[N_];
  __shared__ float sN[N_];
  const int b = blockIdx.x;
  const int tid = threadIdx.x;
  const uint32_t HALF = (B_ * N_) / 2;  // 48000 (threefry counter split)

  float mySP[3], mySN[3];
  bool  myPC[3], myNC[3];
#pragma unroll
  for (int it = 0; it < 3; ++it) {
    int n = tid + it * 512;
    if (n < N_) {
      size_t g = (size_t)b * N_ + n;
      float mv = merged[g];
      bool posC = mv > 0.5f;
      bool negC = (mv < 0.5f) && (mv > 0.1f);
      uint32_t gi = (uint32_t)g;
      uint32_t x0 = (gi < HALF) ? gi : gi - HALF;
      uint32_t x1 = (gi < HALF) ? gi + HALF : gi;
      uint32_t o0, o1;
      threefry2x32(kp0, kp1, x0, x1, o0, o1);
      float uP = bits_to_u01((gi < HALF) ? o0 : o1);
      threefry2x32(kn0, kn1, x0, x1, o0, o1);
      float uN = bits_to_u01((gi < HALF) ? o0 : o1);
      float spv = posC ? uP : 0.0f;
      float snv = negC ? uN : 0.0f;
      sP[n] = spv; sN[n] = snv;
      mySP[it] = spv; mySN[it] = snv; myPC[it] = posC; myNC[it] = negC;
    }
  }
  __syncthreads();

#pragma unroll
  for (int it = 0; it < 3; ++it) {
    int n = tid + it * 512;
    if (n >= N_) continue;
    const float sp = mySP[it], sn = mySN[it];
    int cp = 0, cn = 0;
    for (int j = 0; j < N_; ++j) {
      float vP = sP[j], vN = sN[j];
      cp += (vP > sp) || (vP == sp && j < n);
      cn += (vN > sn) || (vN == sn && j < n);
    }
    bool posSel = myPC[it] && (cp < SEL_);
    bool negSel = myNC[it] && (cn < SEL_);

    size_t g = (size_t)b * N_ + n;
    int mj = maxIdx[g];
    int label = (posSel ? gtl[b * M_ + mj] : -1) + (negSel ? 1 : 0);

    float gy1 = 0.f, gx1 = 0.f, gy2 = 0.f, gx2 = 0.f;
    if (posSel) {
      const float4 gb = ((const float4*)gt)[(size_t)b * M_ + mj];
      gy1 = gb.x; gx1 = gb.y; gy2 = gb.z; gx2 = gb.w;
    }
    const float4 rb = ((const float4*)roi)[g];
    float bh = rb.z - rb.x, bw = rb.w - rb.y;
    float bcy = rb.x + 0.5f * bh, bcx = rb.y + 0.5f * bw;
    float gh = gy2 - gy1, gw = gx2 - gx1;
    float gcy = gy1 + 0.5f * gh, gcx = gx1 + 0.5f * gw;
    float bhs = (bh == 0.0f) ? 1e-3f : bh;
    float bws = (bw == 0.0f) ? 1e-3f : bw;
    float ghs = (gh == 0.0f) ? 1e-3f : gh;
    float gws = (gw == 0.0f) ? 1e-3f : gw;
    float dy = (gh == 0.0f) ? 0.0f : (gcy - bcy) / bhs;
    float dx = (gw == 0.0f) ? 0.0f : (gcx - bcx) / bws;
    float dh = (gh == 0.0f) ? 0.0f : __logf(ghs / bhs);
    float dw = (gw == 0.0f) ? 0.0f : __logf(gws / bws);

    labelOut[g] = label;
    // divide by variances [0.1,0.1,0.2,0.2]
    ((float4*)deltaOut)[g] = make_float4(dy * 10.f, dx * 10.f, dh * 5.f, dw * 5.f);
  }
}

// ---------------------------------------------------------------------------
// Kernel C: streaming one-hot expansion. 155.5 MB of stores -> the bandwidth
// kernel. One wave per (b,n) row; B128 stores for the [81][4] delta row.
// ---------------------------------------------------------------------------
__global__ __launch_bounds__(256) void expand_kernel(
    const int* __restrict__ labelIn, const float* __restrict__ deltaIn,
    float* __restrict__ out) {
  const int lane = threadIdx.x & 31;
  const int wave = threadIdx.x >> 5;
  const long long r = (long long)blockIdx.x * 8 + wave;
  if (r >= (long long)B_ * N_) return;
  const int label = labelIn[r];
  const float4 d = ((const float4*)deltaIn)[r];
  float4* drow = (float4*)(out + (size_t)r * (L_ * 4));
  float*  lrow = out + (size_t)B_ * N_ * L_ * 4 + (size_t)r * L_;
  const float4 z = make_float4(0.f, 0.f, 0.f, 0.f);
#pragma unroll
  for (int k = 0; k < 3; ++k) {
    int c = lane + k * 32;
    if (c < L_) {
      drow[c] = (c == label) ? d : z;        // global_store_b128
      lrow[c] = (c == label) ? 1.0f : 0.0f;  // global_store_b32
    }
  }
}

// ---------------------------------------------------------------------------
extern "C" void kernel_launch(void* const* d_in, const int* in_sizes, int n_in,
                              void* d_out, int out_size, void* d_ws, size_t ws_size,
                              hipStream_t stream) {
  const float* roi = (const float*)d_in[0];   // [64,1500,4] f32
  const float* gt  = (const float*)d_in[1];   // [64,100,4]  f32
  const int*   gtl = (const int*)d_in[2];     // [64,100]    i32

  // Workspace layout (floats): merged | maxIdx | labels | deltas(4x)
  float* wsf    = (float*)d_ws;
  float* merged = wsf;
  int*   maxIdx = (int*)(wsf + (size_t)B_ * N_);
  int*   labels = (int*)(wsf + 2 * (size_t)B_ * N_);
  float* deltas = wsf + 3 * (size_t)B_ * N_;   // 16B-aligned (1,152,000 B)

  // JAX key(42) = (0,42); split -> kp (pos mask key), kn (neg mask key).
  // split counts [0,1,2,3] pair as (0,2) and (1,3); out = [o0,o0',o1,o1'].
  uint32_t p0, p1, q0, q1;
  threefry2x32(0u, 42u, 0u, 2u, p0, q0);  // p0=kp[0], q0=kn[0]
  threefry2x32(0u, 42u, 1u, 3u, p1, q1);  // p1=kp[1], q1=kn[1]

  iou_argmax_kernel<<<B_ * CHUNKS, 256, 0, stream>>>(roi, gt, merged, maxIdx);
  select_delta_kernel<<<B_, 512, 0, stream>>>(roi, gt, gtl, merged, maxIdx,
                                              labels, deltas, p0, p1, q0, q1);
  expand_kernel<<<(B_ * N_) / 8, 256, 0, stream>>>(labels, deltas, (float*)d_out);
}